// BioLayer_64914135711797
// MI455X (gfx1250) — compile-verified
//
#include <hip/hip_runtime.h>
#include <hip/hip_bf16.h>
#include <math.h>

#define BB 64
#define NN 50000
#define EE 1600000
#define LL 20
#define BN_EPS 1e-5f
#define KCHUNK 784       // 64 chunks cover 50176 >= 50000; last chunk length 608 (mult of 4)
#define NCHUNKS 64

typedef __attribute__((ext_vector_type(2))) float v2f;
typedef __attribute__((ext_vector_type(8))) float v8f;

// ---------------- zero workspace (sums + counts) ----------------
__global__ void k_zero(float* __restrict__ p, long n) {
    long i = (long)blockIdx.x * blockDim.x + threadIdx.x;
    long stride = (long)gridDim.x * blockDim.x;
    for (; i < n; i += stride) p[i] = 0.0f;
}

// ---------------- transpose x [B,N] -> xT [N,B], 64x64 tiles via LDS ----------------
__global__ void k_transpose(const float* __restrict__ x, float* __restrict__ xT) {
    __shared__ float tile[64 * 65];
    int n0   = blockIdx.x * 64;
    int col  = threadIdx.x & 63;   // node-local or batch index
    int rowg = threadIdx.x >> 6;   // 0..3
    for (int b = rowg; b < 64; b += 4) {
        int n = n0 + col;
        tile[b * 65 + col] = (n < NN) ? x[(long)b * NN + n] : 0.0f;  // coalesced read
    }
    __syncthreads();
    for (int nl = rowg; nl < 64; nl += 4) {
        int n = n0 + nl;
        if (n < NN) xT[(long)n * 64 + col] = tile[col * 65 + nl];    // coalesced write
    }
}

// ---------------- per-edge affine gather + scatter-add (L2-resident) ----------------
// 64 threads per edge -> alpha/bias/src/dst are wave-uniform under wave32.
__global__ void k_edges(const float* __restrict__ xT, const int* __restrict__ ei,
                        const float* __restrict__ alpha, const float* __restrict__ bias,
                        float* __restrict__ sums, float* __restrict__ counts) {
    long gtid = (long)blockIdx.x * blockDim.x + threadIdx.x;
    long e = gtid >> 6;
    int  t = (int)(gtid & 63);
    if (e >= EE) return;
    int   s  = ei[e];
    int   d  = ei[(long)EE + e];
    float a  = alpha[e];
    float bi = bias[e];
    float val = fmaf(a, xT[(long)s * 64 + t], bi);
    atomicAdd(&sums[(long)d * 64 + t], val);
    if (t == 0) atomicAdd(&counts[d], 1.0f);
}

// ---------------- mean-divide, tanh, batchnorm over B; x_hat written in place ----------------
// One wave per node: 64 batch values as 2 per lane; shfl_xor reduction (wave32).
__global__ void k_finalize(float* __restrict__ sums /* in sums, out x_hat */,
                           const float* __restrict__ counts,
                           const float* __restrict__ gamma, const float* __restrict__ beta,
                           float* __restrict__ bn_out /* [B][N] */) {
    __shared__ float tile[64 * 65];
    int n0   = blockIdx.x * 64;
    int wave = threadIdx.x >> 5;   // 0..7
    int lane = threadIdx.x & 31;
    for (int i = 0; i < 8; ++i) {
        int lnode = wave * 8 + i;
        int n = n0 + lnode;
        if (n < NN) {
            float2 s = *(const float2*)&sums[(long)n * 64 + 2 * lane];
            float inv = 1.0f / fmaxf(counts[n], 1.0f);
            float xh0 = s.x * inv, xh1 = s.y * inv;
            float2 xh; xh.x = xh0; xh.y = xh1;
            *(float2*)&sums[(long)n * 64 + 2 * lane] = xh;   // x_hat in place (for GEMM)
            float t0 = tanhf(xh0), t1 = tanhf(xh1);
            float sm = t0 + t1;
            float sq = t0 * t0 + t1 * t1;
            #pragma unroll
            for (int off = 16; off >= 1; off >>= 1) {
                sm += __shfl_xor(sm, off, 32);
                sq += __shfl_xor(sq, off, 32);
            }
            float mu = sm * (1.0f / 64.0f);
            float va = sq * (1.0f / 64.0f) - mu * mu;
            float rs = rsqrtf(va + BN_EPS);
            float g = gamma[n], be = beta[n];
            tile[lnode * 65 + 2 * lane]     = (t0 - mu) * rs * g + be;
            tile[lnode * 65 + 2 * lane + 1] = (t1 - mu) * rs * g + be;
        }
    }
    __syncthreads();
    int col  = threadIdx.x & 63;   // node-local
    int rowg = threadIdx.x >> 6;   // 0..3
    int n = n0 + col;
    if (n < NN) {
        for (int b = rowg; b < 64; b += 4)
            bn_out[(long)b * NN + n] = tile[col * 65 + b];   // coalesced over n
    }
}

// ---------------- split-K GEMM via V_WMMA_F32_16X16X4_F32 ----------------
// pred_part[c] = x_hat[64, Kc] @ W.T[Kc, 20]; 8 waves = 4 M-tiles x 2 N-tiles.
// Unrolled x4 so 16 loads are in flight per 4 WMMAs (latency hiding).
__global__ void k_gemm_part(const float* __restrict__ xhatT /* [N,64] */,
                            const float* __restrict__ W     /* [L,N]  */,
                            float* __restrict__ part) {
    int c    = blockIdx.x;
    int wave = threadIdx.x >> 5;
    int lane = threadIdx.x & 31;
    int half = lane >> 4;          // 0: K+0/1, 1: K+2/3 (A & B lane split)
    int l16  = lane & 15;
    int m_tile = wave >> 1;        // 0..3 -> batch rows
    int n_tile = wave & 1;         // 0..1 -> label cols
    int m    = m_tile * 16 + l16;
    int lcol = n_tile * 16 + l16;
    int wrow = (lcol < LL) ? lcol : 0;      // clamp; masked at store
    const float* Wrow = W + (long)wrow * NN;

    long kbeg = (long)c * KCHUNK;
    long kend = kbeg + KCHUNK; if (kend > NN) kend = NN;

    v8f acc = {};
    #pragma unroll 4
    for (long k = kbeg; k < kend; k += 4) {
        long k0 = k + half * 2;
        long k1 = k0 + 1;
        v2f a, b;
        a.x = xhatT[k0 * 64 + m];
        a.y = xhatT[k1 * 64 + m];
        b.x = Wrow[k0];
        b.y = Wrow[k1];
        acc = __builtin_amdgcn_wmma_f32_16x16x4_f32(
            /*neg_a=*/false, a, /*neg_b=*/false, b,
            /*c_mod=*/(short)0, acc, /*reuse_a=*/false, /*reuse_b=*/false);
    }

    float* dst = part + (long)c * (BB * LL);
    #pragma unroll
    for (int i = 0; i < 8; ++i) {
        int mrow = m_tile * 16 + half * 8 + i;  // D layout: lanes 16-31 hold M=8..15
        int ncol = n_tile * 16 + l16;
        if (ncol < LL) dst[mrow * LL + ncol] = acc[i];
    }
}

// ---------------- deterministic reduction of split-K partials + bias ----------------
__global__ void k_reduce_pred(const float* __restrict__ part, const float* __restrict__ b_lin,
                              float* __restrict__ pred) {
    int i = blockIdx.x * blockDim.x + threadIdx.x;
    if (i >= BB * LL) return;
    float s = b_lin[i % LL];
    for (int c = 0; c < NCHUNKS; ++c) s += part[c * (BB * LL) + i];
    pred[i] = s;
}

extern "C" void kernel_launch(void* const* d_in, const int* in_sizes, int n_in,
                              void* d_out, int out_size, void* d_ws, size_t ws_size,
                              hipStream_t stream) {
    const float* x     = (const float*)d_in[0];
    const int*   ei    = (const int*)  d_in[1];
    const float* alpha = (const float*)d_in[2];
    const float* bias  = (const float*)d_in[3];
    const float* W     = (const float*)d_in[4];
    const float* b_lin = (const float*)d_in[5];
    const float* gamma = (const float*)d_in[6];
    const float* beta  = (const float*)d_in[7];
    float* out = (float*)d_out;          // [0,1280): pred ; [1280, 1280+64*50000): bn

    float* ws     = (float*)d_ws;
    float* sums   = ws;                          // NN*64 floats (becomes x_hat in place)
    float* counts = sums + (long)NN * 64;        // NN floats
    float* xT     = counts + NN;                 // NN*64 floats
    float* part   = xT + (long)NN * 64;          // NCHUNKS*BB*LL floats

    // 1) zero accumulators (sums + counts are contiguous)
    k_zero<<<4096, 256, 0, stream>>>(sums, (long)NN * 64 + NN);
    // 2) x [B,N] -> xT [N,B]
    k_transpose<<<(NN + 63) / 64, 256, 0, stream>>>(x, xT);
    // 3) per-edge affine + scatter-add (L2 resident)
    long edge_threads = (long)EE * 64;
    k_edges<<<(int)((edge_threads + 255) / 256), 256, 0, stream>>>(xT, ei, alpha, bias, sums, counts);
    // 4) mean/tanh/batchnorm (writes bn, leaves x_hat in `sums`)
    k_finalize<<<(NN + 63) / 64, 256, 0, stream>>>(sums, counts, gamma, beta, out + BB * LL);
    // 5) split-K WMMA GEMM partials
    k_gemm_part<<<NCHUNKS, 256, 0, stream>>>(sums, W, part);
    // 6) deterministic partial reduction + bias -> pred
    k_reduce_pred<<<(BB * LL + 255) / 256, 256, 0, stream>>>(part, b_lin, out);
}